// EGATE_9603546874323
// MI455X (gfx1250) — compile-verified
//
#include <hip/hip_runtime.h>
#include <stdint.h>
#include <math.h>

#define N_NODES 10000
#define E_EDGES 320000
#define E2      (E_EDGES + N_NODES)   // edges + self loops = 330000
#define IN_CH   128
#define HIDC    128
#define OUTC    64
#define HEADS   3
#define KCL     10
#define HC1     (HEADS * HIDC)        // 384
#define HC2     (HEADS * OUTC)        // 192
#define NEG_SLOPE 0.2f

typedef float v2f __attribute__((ext_vector_type(2)));
typedef float v8f __attribute__((ext_vector_type(8)));

// ---------------------------------------------------------------------------
// Generic fp32 WMMA GEMM: C[M,Nc] = A[M,K] * B, one 16x16 tile per wave32,
// V_WMMA_F32_16X16X4_F32, K stepped by 4. CLEAN_A: nan_to_num on A loads.
// Used for the small feature GEMMs (x@W1, h1@W2).
// ---------------------------------------------------------------------------
template <bool CLEAN_A>
__global__ void gemm_wmma_f32(const float* __restrict__ A,
                              const float* __restrict__ B,
                              float* __restrict__ C,
                              int K, int lda, int ldb, int ldc,
                              int tilesN, int totalTiles) {
    int w = blockIdx.x * (blockDim.x >> 5) + (threadIdx.x >> 5);
    if (w >= totalTiles) return;                 // wave-uniform exit
    const int tM   = w / tilesN;
    const int tN   = w % tilesN;
    const int lane = threadIdx.x & 31;
    const int r16  = lane & 15;                  // A row / C col within tile
    const int kh   = lane >> 4;                  // K-half select (0 or 1)

    const float* Arow = A + (size_t)(tM * 16 + r16) * lda;
    const int    bcol = tN * 16 + r16;

    v8f c = {};
    for (int k0 = 0; k0 < K; k0 += 4) {
        const int ka = k0 + kh * 2;
        v2f a, b;
        a.x = Arow[ka];
        a.y = Arow[ka + 1];
        if (CLEAN_A) {
            a.x = (a.x != a.x) ? 0.0f : a.x;
            a.y = (a.y != a.y) ? 0.0f : a.y;
        }
        b.x = B[(size_t)ka * ldb + bcol];
        b.y = B[(size_t)(ka + 1) * ldb + bcol];
        // 8 args: (neg_a, A, neg_b, B, c_mod, C, reuse_a, reuse_b)
        c = __builtin_amdgcn_wmma_f32_16x16x4_f32(
                false, a, false, b, (short)0, c, false, false);
    }

    #pragma unroll
    for (int r = 0; r < 8; ++r) {
        const int row = tM * 16 + r + kh * 8;    // C layout: VGPR r -> M=r / M=r+8
        C[(size_t)row * ldc + bcol] = c[r];
    }
}

// ---------------------------------------------------------------------------
// A_hat = z @ z^T specialized kernel. 16x80 output per wave (5 WMMA tiles,
// N = 10000 = 125*80 and 625*16 exactly, so no tails). Per k-step:
// 1 A-load + 5 B-loads feed 5 v_wmma -> ~1.7x fewer loads than 1x1 tiling.
// Result (400MB > 192MB L2) is stored non-temporally so z stays L2-resident.
// ---------------------------------------------------------------------------
#define NB 5
__global__ void zzT_wmma(const float* __restrict__ z, float* __restrict__ C) {
    const int tilesN = N_NODES / (NB * 16);               // 125
    const int total  = (N_NODES / 16) * tilesN;           // 78125
    int w = blockIdx.x * (blockDim.x >> 5) + (threadIdx.x >> 5);
    if (w >= total) return;                               // wave-uniform exit
    const int tM   = w / tilesN;
    const int tN   = w % tilesN;
    const int lane = threadIdx.x & 31;
    const int r16  = lane & 15;
    const int kh   = lane >> 4;

    const float* Arow = z + (size_t)(tM * 16 + r16) * HC2;
    const float* Brow[NB];
    #pragma unroll
    for (int j = 0; j < NB; ++j)
        Brow[j] = z + (size_t)(tN * (NB * 16) + j * 16 + r16) * HC2;

    v8f acc[NB] = {};
    for (int k0 = 0; k0 < HC2; k0 += 4) {
        const int ka = k0 + kh * 2;
        v2f a;
        a.x = Arow[ka];
        a.y = Arow[ka + 1];
        #pragma unroll
        for (int j = 0; j < NB; ++j) {
            v2f b;
            b.x = Brow[j][ka];
            b.y = Brow[j][ka + 1];
            acc[j] = __builtin_amdgcn_wmma_f32_16x16x4_f32(
                         false, a, false, b, (short)0, acc[j], false, false);
        }
    }

    #pragma unroll
    for (int j = 0; j < NB; ++j) {
        const int col = tN * (NB * 16) + j * 16 + r16;
        #pragma unroll
        for (int r = 0; r < 8; ++r) {
            const int row = tM * 16 + r + kh * 8;
            __builtin_nontemporal_store(acc[j][r],
                                        &C[(size_t)row * N_NODES + col]);
        }
    }
}

// ---------------------------------------------------------------------------
// Edge preprocessing: int64 edge_index -> int32 src/dst, append self loops.
// ---------------------------------------------------------------------------
__global__ void prep_edges(const long long* __restrict__ ei,
                           int* __restrict__ src32, int* __restrict__ dst32) {
    int e = blockIdx.x * blockDim.x + threadIdx.x;
    if (e >= E2) return;
    if (e < E_EDGES) {
        src32[e] = (int)ei[e];
        dst32[e] = (int)ei[E_EDGES + e];
    } else {
        int n = e - E_EDGES;
        src32[e] = n;
        dst32[e] = n;
    }
}

// a_s[n,h] = <h_lin[n,h,:], att_s[h,:]>, same for a_d. One thread per (n,h).
__global__ void node_attn(const float* __restrict__ h,
                          const float* __restrict__ att_s,
                          const float* __restrict__ att_d,
                          float* __restrict__ a_s, float* __restrict__ a_d,
                          int C) {
    int i = blockIdx.x * blockDim.x + threadIdx.x;
    if (i >= N_NODES * HEADS) return;
    int n = i / HEADS, hh = i % HEADS;
    const float* hp = h + (size_t)n * HEADS * C + (size_t)hh * C;
    const float* as = att_s + hh * C;
    const float* ad = att_d + hh * C;
    float ss = 0.0f, sd = 0.0f;
    for (int c = 0; c < C; ++c) { float v = hp[c]; ss += v * as[c]; sd += v * ad[c]; }
    a_s[i] = ss;
    a_d[i] = sd;
}

// EDIM==1 collapses the edge-feature path: a_e[e,h] = eattr[e] * s_e[h].
__global__ void edge_scale(const float* __restrict__ We,
                           const float* __restrict__ att_e,
                           float* __restrict__ s_e, int C) {
    int hh = threadIdx.x;
    if (hh >= HEADS) return;
    float s = 0.0f;
    for (int c = 0; c < C; ++c) s += We[hh * C + c] * att_e[hh * C + c];
    s_e[hh] = s;
}

__global__ void fill_f32(float* __restrict__ p, float v, int n) {
    int i = blockIdx.x * blockDim.x + threadIdx.x;
    if (i < n) p[i] = v;
}

__device__ __forceinline__ float atomicMaxF(float* addr, float val) {
    int* ai = (int*)addr;
    int old = __float_as_int(*addr);
    while (__int_as_float(old) < val) {
        int assumed = old;
        old = atomicCAS(ai, assumed, __float_as_int(val));
        if (old == assumed) break;
    }
    return __int_as_float(old);
}

// alpha = leaky_relu(a_s[src] + a_d[dst] + eattr*s_e) ; segment max into m.
__global__ void edge_alpha(const int* __restrict__ src, const int* __restrict__ dst,
                           const float* __restrict__ eattr,
                           const float* __restrict__ a_s, const float* __restrict__ a_d,
                           const float* __restrict__ s_e,
                           float* __restrict__ alpha, float* __restrict__ m) {
    int i = blockIdx.x * blockDim.x + threadIdx.x;
    if (i >= E2 * HEADS) return;
    int e = i / HEADS, hh = i % HEADS;
    float av = (e < E_EDGES) ? eattr[e] : 1.0f;   // self-loop fill_value = 1
    float v = a_s[src[e] * HEADS + hh] + a_d[dst[e] * HEADS + hh] + av * s_e[hh];
    v = (v > 0.0f) ? v : NEG_SLOPE * v;
    alpha[i] = v;
    atomicMaxF(&m[dst[e] * HEADS + hh], v);
}

// ex = exp(alpha - m[dst]) in place; segment sum into denom.
__global__ void edge_exp(const int* __restrict__ dst,
                         const float* __restrict__ m,
                         float* __restrict__ alpha, float* __restrict__ denom) {
    int i = blockIdx.x * blockDim.x + threadIdx.x;
    if (i >= E2 * HEADS) return;
    int e = i / HEADS, hh = i % HEADS;
    float ex = __expf(alpha[i] - m[dst[e] * HEADS + hh]);
    alpha[i] = ex;
    atomicAdd(&denom[dst[e] * HEADS + hh], ex);
}

// agg[dst] += h_lin[src] * coef  (one block per edge, HEADS*C threads).
__global__ void aggregate(const int* __restrict__ src, const int* __restrict__ dst,
                          const float* __restrict__ ex, const float* __restrict__ denom,
                          const float* __restrict__ h, float* __restrict__ agg,
                          int C) {
    int e = blockIdx.x;
    int c = threadIdx.x;                 // 0 .. HEADS*C-1
    int hh = c / C;
    int s = src[e], d = dst[e];
    float coef = ex[e * HEADS + hh] / (denom[d * HEADS + hh] + 1e-16f);
    atomicAdd(&agg[(size_t)d * (HEADS * C) + c],
              h[(size_t)s * (HEADS * C) + c] * coef);
}

// out = agg + b, optional ELU.
__global__ void bias_act(float* __restrict__ buf, const float* __restrict__ b,
                         int HC, int elu) {
    int i = blockIdx.x * blockDim.x + threadIdx.x;
    if (i >= N_NODES * HC) return;
    float v = buf[i] + b[i % HC];
    if (elu) v = (v > 0.0f) ? v : (__expf(v) - 1.0f);
    buf[i] = v;
}

// z = h2 / max(||h2||, 1e-12), one thread per node.
__global__ void normalize_z(const float* __restrict__ h2, float* __restrict__ z) {
    int n = blockIdx.x * blockDim.x + threadIdx.x;
    if (n >= N_NODES) return;
    const float* r = h2 + (size_t)n * HC2;
    float s = 0.0f;
    for (int c = 0; c < HC2; ++c) s += r[c] * r[c];
    float inv = 1.0f / fmaxf(sqrtf(s), 1e-12f);
    float* zo = z + (size_t)n * HC2;
    for (int c = 0; c < HC2; ++c) zo[c] = r[c] * inv;
}

// p = softmax(z @ Wp + bp) over K=10, one thread per node.
__global__ void cluster_p(const float* __restrict__ z, const float* __restrict__ Wp,
                          const float* __restrict__ bp, float* __restrict__ p) {
    int n = blockIdx.x * blockDim.x + threadIdx.x;
    if (n >= N_NODES) return;
    const float* zr = z + (size_t)n * HC2;
    float lg[KCL];
    for (int k = 0; k < KCL; ++k) {
        float s = bp[k];
        for (int c = 0; c < HC2; ++c) s += zr[c] * Wp[c * KCL + k];
        lg[k] = s;
    }
    float mx = lg[0];
    for (int k = 1; k < KCL; ++k) mx = fmaxf(mx, lg[k]);
    float den = 0.0f;
    for (int k = 0; k < KCL; ++k) { lg[k] = __expf(lg[k] - mx); den += lg[k]; }
    for (int k = 0; k < KCL; ++k) p[(size_t)n * KCL + k] = lg[k] / den;
}

// ---------------------------------------------------------------------------
extern "C" void kernel_launch(void* const* d_in, const int* in_sizes, int n_in,
                              void* d_out, int out_size, void* d_ws, size_t ws_size,
                              hipStream_t stream) {
    const float*     x     = (const float*)d_in[0];
    const long long* ei    = (const long long*)d_in[1];
    const float*     eattr = (const float*)d_in[2];
    const float*     W1    = (const float*)d_in[3];
    const float*     atts1 = (const float*)d_in[4];
    const float*     attd1 = (const float*)d_in[5];
    const float*     We1   = (const float*)d_in[6];
    const float*     atte1 = (const float*)d_in[7];
    const float*     b1    = (const float*)d_in[8];
    const float*     W2    = (const float*)d_in[9];
    const float*     atts2 = (const float*)d_in[10];
    const float*     attd2 = (const float*)d_in[11];
    const float*     We2   = (const float*)d_in[12];
    const float*     atte2 = (const float*)d_in[13];
    const float*     b2    = (const float*)d_in[14];
    const float*     Wp    = (const float*)d_in[15];
    const float*     bp    = (const float*)d_in[16];

    // Output layout: A_hat [N,N], p [N,K], z [N,HC2]
    float* A_hat = (float*)d_out;
    float* p_out = A_hat + (size_t)N_NODES * N_NODES;
    float* z_out = p_out + (size_t)N_NODES * KCL;

    // Workspace layout
    float* f     = (float*)d_ws;
    size_t o     = 0;
    float* bufA  = f + o; o += (size_t)N_NODES * HC1;   // linear features
    float* bufB  = f + o; o += (size_t)N_NODES * HC1;   // aggregation / activations
    float* a_s   = f + o; o += (size_t)N_NODES * HEADS;
    float* a_d   = f + o; o += (size_t)N_NODES * HEADS;
    float* mbuf  = f + o; o += (size_t)N_NODES * HEADS;
    float* dbuf  = f + o; o += (size_t)N_NODES * HEADS;
    float* alpha = f + o; o += (size_t)E2 * HEADS;
    float* s_e   = f + o; o += 16;
    int*   src32 = (int*)(f + o); o += E2;
    int*   dst32 = (int*)(f + o); o += E2;

    const int NH = N_NODES * HEADS;
    const int EH = E2 * HEADS;

    // ---- shared edge preprocessing ----
    prep_edges<<<(E2 + 255) / 256, 256, 0, stream>>>(ei, src32, dst32);

    // ================= GAT layer 1 =================
    {   // h_lin1 = nan_to_num(x) @ W1   [N,384]
        int tilesN = HC1 / 16, total = (N_NODES / 16) * tilesN;
        gemm_wmma_f32<true><<<(total + 7) / 8, 256, 0, stream>>>(
            x, W1, bufA, IN_CH, IN_CH, HC1, HC1, tilesN, total);
    }
    node_attn<<<(NH + 255) / 256, 256, 0, stream>>>(bufA, atts1, attd1, a_s, a_d, HIDC);
    edge_scale<<<1, 32, 0, stream>>>(We1, atte1, s_e, HIDC);
    fill_f32<<<(NH + 255) / 256, 256, 0, stream>>>(mbuf, -1e30f, NH);
    hipMemsetAsync(dbuf, 0, (size_t)NH * sizeof(float), stream);
    edge_alpha<<<(EH + 255) / 256, 256, 0, stream>>>(src32, dst32, eattr, a_s, a_d, s_e, alpha, mbuf);
    edge_exp<<<(EH + 255) / 256, 256, 0, stream>>>(dst32, mbuf, alpha, dbuf);
    hipMemsetAsync(bufB, 0, (size_t)N_NODES * HC1 * sizeof(float), stream);
    aggregate<<<E2, HC1, 0, stream>>>(src32, dst32, alpha, dbuf, bufA, bufB, HIDC);
    bias_act<<<((N_NODES * HC1) + 255) / 256, 256, 0, stream>>>(bufB, b1, HC1, 1);

    // ================= GAT layer 2 =================
    {   // h_lin2 = h1act @ W2   [N,192]
        int tilesN = HC2 / 16, total = (N_NODES / 16) * tilesN;
        gemm_wmma_f32<false><<<(total + 7) / 8, 256, 0, stream>>>(
            bufB, W2, bufA, HC1, HC1, HC2, HC2, tilesN, total);
    }
    node_attn<<<(NH + 255) / 256, 256, 0, stream>>>(bufA, atts2, attd2, a_s, a_d, OUTC);
    edge_scale<<<1, 32, 0, stream>>>(We2, atte2, s_e, OUTC);
    fill_f32<<<(NH + 255) / 256, 256, 0, stream>>>(mbuf, -1e30f, NH);
    hipMemsetAsync(dbuf, 0, (size_t)NH * sizeof(float), stream);
    edge_alpha<<<(EH + 255) / 256, 256, 0, stream>>>(src32, dst32, eattr, a_s, a_d, s_e, alpha, mbuf);
    edge_exp<<<(EH + 255) / 256, 256, 0, stream>>>(dst32, mbuf, alpha, dbuf);
    hipMemsetAsync(bufB, 0, (size_t)N_NODES * HC2 * sizeof(float), stream);
    aggregate<<<E2, HC2, 0, stream>>>(src32, dst32, alpha, dbuf, bufA, bufB, OUTC);
    bias_act<<<((N_NODES * HC2) + 255) / 256, 256, 0, stream>>>(bufB, b2, HC2, 0);

    // ================= heads =================
    normalize_z<<<(N_NODES + 255) / 256, 256, 0, stream>>>(bufB, z_out);
    cluster_p<<<(N_NODES + 255) / 256, 256, 0, stream>>>(z_out, Wp, bp, p_out);

    // A_hat = z @ z^T : 16x80 per wave, fp32 WMMA, non-temporal stores.
    {
        int total = (N_NODES / 16) * (N_NODES / (NB * 16));   // 78125 wave-tiles
        zzT_wmma<<<(total + 7) / 8, 256, 0, stream>>>(z_out, A_hat);
    }
}